// GraphAttentionHeadModule_52467320488303
// MI455X (gfx1250) — compile-verified
//
#include <hip/hip_runtime.h>
#include <math.h>

typedef float v2f __attribute__((ext_vector_type(2)));
typedef float v8f __attribute__((ext_vector_type(8)));

#define NN 2048
#define DIN 256
#define DOUT 64
#define ALPHA 0.2f
#define LN_EPS 1e-5f

// ---------------------------------------------------------------------------
// CDNA5 async global->LDS copy (16B), tracked by ASYNCcnt (ISA cdna5 §08).
// GV addressing: 64-bit global addr in VGPR pair, LDS byte offset in VDST VGPR
// (low 32 bits of the generic shared-aperture pointer == LDS offset).
// ---------------------------------------------------------------------------
__device__ __forceinline__ void async_ld_b128(const void* gptr, void* lptr) {
  const unsigned int l = (unsigned int)(unsigned long long)(uintptr_t)lptr;
  const unsigned long long g = (unsigned long long)(uintptr_t)gptr;
  asm volatile("global_load_async_to_lds_b128 %0, %1, off"
               :: "v"(l), "v"(g) : "memory");
}
__device__ __forceinline__ void wait_async0() {
  asm volatile("s_wait_asynccnt 0x0" ::: "memory");
}

// ---------------------------------------------------------------------------
// Kernel 1: h = LayerNorm(features @ phi_w^T + phi_b);  s = h @ mu_w
// Grid: 128 blocks (16 rows each), 128 threads (4 waves, one 16x16 tile each).
// ---------------------------------------------------------------------------
__global__ __launch_bounds__(128) void gat_h_kernel(
    const float* __restrict__ feat, const float* __restrict__ phi_w,
    const float* __restrict__ phi_b, const float* __restrict__ ln_g,
    const float* __restrict__ ln_b, const float* __restrict__ mu_w,
    float* __restrict__ h_out, float* __restrict__ s_out) {
  __shared__ float phi_s[64][132];    // padded: conflict-free frag reads
  __shared__ float feat_s[16][132];
  __shared__ float h_s[16][68];
  const int t = threadIdx.x;
  const int wave = t >> 5, lane = t & 31;
  const int m = lane & 15, hi = lane >> 4;   // WMMA lane decomposition
  const int n0 = wave * 16;                  // output-col tile of this wave
  const int i0 = blockIdx.x * 16;            // output-row tile of this block

  v8f c = {};
  for (int kh = 0; kh < 2; ++kh) {           // split K=256 into 2x128 (LDS budget)
    const int kb = kh * 128;
    __syncthreads();
    // async-stage phi_w[64][128] chunk: DMA straight into LDS, no VGPR bounce
    for (int u = t; u < 64 * 32; u += 128) {
      const int r = u >> 5, q = u & 31;
      async_ld_b128(&phi_w[r * DIN + kb + q * 4], &phi_s[r][q * 4]);
    }
    // async-stage features[16][128] chunk
    for (int u = t; u < 16 * 32; u += 128) {
      const int r = u >> 5, q = u & 31;
      async_ld_b128(&feat[(size_t)(i0 + r) * DIN + kb + q * 4],
                    &feat_s[r][q * 4]);
    }
    wait_async0();
    __syncthreads();
#pragma unroll
    for (int k = 0; k < 128; k += 4) {
      // A (16x4 f32): lane m = M, VGPR v -> K = k + v + 2*hi  (=> one ds_load_b64)
      v2f a = *(const v2f*)&feat_s[m][k + 2 * hi];
      // B (4x16):   B[k][n] = phi_w[n][k]
      v2f b = *(const v2f*)&phi_s[n0 + m][k + 2 * hi];
      c = __builtin_amdgcn_wmma_f32_16x16x4_f32(false, a, false, b, (short)0, c,
                                                false, false);
    }
  }
  // C layout: VGPR r -> row M = r + 8*hi, col N = n0 + m
  const float bias = phi_b[n0 + m];
#pragma unroll
  for (int r = 0; r < 8; ++r) h_s[r + 8 * hi][n0 + m] = c[r] + bias;
  __syncthreads();

  if (t < 16) {  // LayerNorm row t + projection s = h . mu_w
    float mu = 0.f;
    for (int d = 0; d < DOUT; ++d) mu += h_s[t][d];
    mu *= (1.0f / DOUT);
    float var = 0.f;
    for (int d = 0; d < DOUT; ++d) { float x = h_s[t][d] - mu; var += x * x; }
    var *= (1.0f / DOUT);
    const float rstd = rsqrtf(var + LN_EPS);
    float sacc = 0.f;
    const size_t row = (size_t)(i0 + t) * DOUT;
    for (int d = 0; d < DOUT; ++d) {
      float v = (h_s[t][d] - mu) * rstd * ln_g[d] + ln_b[d];
      h_out[row + d] = v;
      sacc += v * mu_w[d];
    }
    s_out[i0 + t] = sacc;
  }
}

// ---------------------------------------------------------------------------
// Kernel 2: e[i,j] = lrelu(s_i + s_j + mu_b), masked softmax over j,
//           out = elu(a @ h).  Flash-style: a never hits global memory.
// Async h-chunk DMA overlapped with attention-tile VALU work each iteration.
// Grid: 128 blocks (16 rows each), 128 threads (4 waves -> 4 col tiles of 16).
// ---------------------------------------------------------------------------
__global__ __launch_bounds__(128) void gat_attn_kernel(
    const int* __restrict__ adj, const float* __restrict__ h,
    const float* __restrict__ s, const float* __restrict__ mu_b_p,
    float* __restrict__ out) {
  __shared__ float s_lds[NN];          // 8 KB: all scores staged once
  __shared__ float h_tile[64][64];     // 16 KB: h chunk for WMMA B
  __shared__ float a_tile[16][68];     // padded attention tile (WMMA A)
  __shared__ float pmax[16][8], psum[16][8];
  __shared__ float rmax[16], rinv[16];
  const int t = threadIdx.x;
  const int wave = t >> 5, lane = t & 31;
  const int m = lane & 15, hi = lane >> 4;
  const int n0 = wave * 16;
  const int i0 = blockIdx.x * 16;

  for (int u = t; u < NN / 4; u += 128)   // async-stage s[2048] into LDS
    async_ld_b128(&s[u * 4], &s_lds[u * 4]);
  wait_async0();
  __syncthreads();

  const float mub = mu_b_p[0];
  const int r = t >> 3, cc = t & 7;           // 8 threads per row
  const float si = s_lds[i0 + r];
  const int* __restrict__ arow = &adj[(size_t)(i0 + r) * NN];

  // Pass 1: per-row online max / sum of exp over masked neighbors
  float mx = -INFINITY, sm = 0.f;
  for (int j = cc; j < NN; j += 8) {
    if (arow[j] != 0) {
      float e = si + s_lds[j] + mub;
      e = e > 0.f ? e : ALPHA * e;
      if (e > mx) { sm *= __expf(mx - e); mx = e; }
      sm += __expf(e - mx);
    }
  }
  pmax[r][cc] = mx;
  psum[r][cc] = sm;
  __syncthreads();
  if (t < 16) {
    float M = -INFINITY, S = 0.f;
    for (int q = 0; q < 8; ++q) {
      const float pm = pmax[t][q], ps = psum[t][q];
      if (pm > M)      { S = S * __expf(M - pm) + ps; M = pm; }
      else if (pm != -INFINITY) S += ps * __expf(pm - M);
    }
    rmax[t] = M;
    rinv[t] = 1.0f / S;   // self-loops guarantee S > 0
  }

  // Pass 2: out_tile += a_tile(16x64) @ h_chunk(64x64), 32 chunks
  v8f c = {};
  for (int j0 = 0; j0 < NN; j0 += 64) {
    __syncthreads();  // also covers rmax/rinv visibility on first iteration
    // (1) kick off async DMA of h[j0:j0+64][0:64] into LDS
    for (int u = t; u < 1024; u += 128) {
      const int rr = u >> 4, q = u & 15;
      async_ld_b128(&h[(size_t)(j0 + rr) * DOUT + q * 4], &h_tile[rr][q * 4]);
    }
    if (j0 + 64 < NN) __builtin_prefetch(arow + j0 + 64, 0, 0);
    // (2) overlap: build normalized attention tile while DMA is in flight
    const float rm = rmax[r], ri = rinv[r];
#pragma unroll
    for (int u = 0; u < 8; ++u) {
      const int jc = cc + 8 * u;
      float a = 0.f;
      if (arow[j0 + jc] != 0) {
        float e = si + s_lds[j0 + jc] + mub;
        e = e > 0.f ? e : ALPHA * e;
        a = __expf(e - rm) * ri;
      }
      a_tile[r][jc] = a;
    }
    // (3) drain DMA, publish to all waves
    wait_async0();
    __syncthreads();
#pragma unroll
    for (int k = 0; k < 64; k += 4) {
      v2f a = *(const v2f*)&a_tile[m][k + 2 * hi];
      v2f b;
      b.x = h_tile[k + 2 * hi][n0 + m];     // B[k][n] = h[j0+k][n]
      b.y = h_tile[k + 1 + 2 * hi][n0 + m];
      c = __builtin_amdgcn_wmma_f32_16x16x4_f32(false, a, false, b, (short)0, c,
                                                false, false);
    }
  }
  // Epilogue: ELU, store
#pragma unroll
  for (int rv = 0; rv < 8; ++rv) {
    float x = c[rv];
    x = x > 0.f ? x : (__expf(x) - 1.0f);
    out[(size_t)(i0 + rv + 8 * hi) * DOUT + n0 + m] = x;
  }
}

extern "C" void kernel_launch(void* const* d_in, const int* in_sizes, int n_in,
                              void* d_out, int out_size, void* d_ws, size_t ws_size,
                              hipStream_t stream) {
  const float* feat  = (const float*)d_in[0];
  const int*   adj   = (const int*)d_in[1];
  const float* phi_w = (const float*)d_in[2];
  const float* phi_b = (const float*)d_in[3];
  const float* ln_g  = (const float*)d_in[4];
  const float* ln_b  = (const float*)d_in[5];
  const float* mu_w  = (const float*)d_in[6];
  const float* mu_b  = (const float*)d_in[7];
  float* out  = (float*)d_out;
  float* h_ws = (float*)d_ws;                        // 2048*64 floats
  float* s_ws = h_ws + (size_t)NN * DOUT;            // 2048 floats

  gat_h_kernel<<<NN / 16, 128, 0, stream>>>(feat, phi_w, phi_b, ln_g, ln_b,
                                            mu_w, h_ws, s_ws);
  gat_attn_kernel<<<NN / 16, 128, 0, stream>>>(adj, h_ws, s_ws, mu_b, out);
}